// TreeTrackBuilder_22119081574555
// MI455X (gfx1250) — compile-verified
//
#include <hip/hip_runtime.h>
#include <hip/hip_bf16.h>
#include <stdint.h>

// ---------------------------------------------------------------------------
// TreeTrackBuilder fused kernel for MI455X (gfx1250, wave32, WMMA).
//
// Roofline: track_h output = 10*20000*512 f32 = 409.6 MB stores -> ~21us floor
// at 23.3 TB/s. GEMMs (~12 GFLOP after fusing Q@K^T into Wqk) are noise, so
// the kernel is organized around coalesced nontemporal b128 stores, f32 WMMA
// (16x16x4) from an LDS-staged x tile, and async global->LDS staging.
// ---------------------------------------------------------------------------

#define N_ROWS 20000
#define FEAT   512
#define CLS    10
#define HEADS  8
#define HID    64
#define HD     512            // HEADS*HID
#define SC     80             // HEADS*CLS (score columns)
#define MT     64             // rows of n per block
#define BLK    512            // threads per block (16 waves)
#define KSTEPS (FEAT / 4)     // K chunks of 4 for wmma f32 16x16x4

typedef __attribute__((ext_vector_type(2))) float v2f;
typedef __attribute__((ext_vector_type(4))) float v4f;
typedef __attribute__((ext_vector_type(8))) float v8f;

// ---------------- prep 1: K[c][hd] = prototype[c,:] @ Wk + bk --------------
__global__ void prep_k_kernel(const float* __restrict__ proto,
                              const float* __restrict__ Wk,
                              const float* __restrict__ bk,
                              float* __restrict__ Kws) {
  const int hd = threadIdx.x;       // 0..511
  const int c  = blockIdx.x;        // 0..9
  const float* pr = proto + c * FEAT;
  float acc = bk[hd];
  for (int f = 0; f < FEAT; ++f)
    acc = fmaf(pr[f], Wk[(size_t)f * HD + hd], acc);
  Kws[c * HD + hd] = acc;
}

// ------- prep 2: Wqk[f][h*10+c] = sum_d Wq[f][h*64+d] * K[c][h*64+d] -------
__global__ void prep_wqk_kernel(const float* __restrict__ Wq,
                                const float* __restrict__ bq,
                                const float* __restrict__ Kws,
                                float* __restrict__ Wqk,
                                float* __restrict__ bqk) {
  __shared__ float Kl[CLS * HD];   // 20 KB
  __shared__ float Ql[8 * HD];     // 16 KB, 8 f-rows per block
  const int tid = threadIdx.x;
  const int f0  = blockIdx.x * 8;
  for (int i = tid; i < CLS * HD; i += BLK) Kl[i] = Kws[i];
  for (int i = tid; i < 8 * HD; i += BLK)  Ql[i] = Wq[(size_t)f0 * HD + i];
  __syncthreads();
  for (int o = tid; o < 8 * SC; o += BLK) {
    const int fr = o / SC, hc = o % SC;
    const int h = hc / CLS, c = hc % CLS;
    const float* q = Ql + fr * HD + h * HID;
    const float* k = Kl + c * HD + h * HID;
    float acc = 0.f;
    for (int d = 0; d < HID; ++d) acc = fmaf(q[d], k[d], acc);
    Wqk[(f0 + fr) * SC + hc] = acc;
  }
  if (blockIdx.x == 0 && tid < SC) {
    const int h = tid / CLS, c = tid % CLS;
    float acc = 0.f;
    for (int d = 0; d < HID; ++d)
      acc = fmaf(bq[h * HID + d], Kl[c * HD + h * HID + d], acc);
    bqk[tid] = acc;
  }
}

// ---------------------------- main fused kernel ----------------------------
__global__ void __launch_bounds__(BLK)
track_kernel(const float* __restrict__ x,
             const unsigned char* __restrict__ mid_mask,
             const float* __restrict__ Wv,
             const float* __restrict__ bv,
             const float* __restrict__ Wqk,
             const float* __restrict__ bqk,
             float* __restrict__ track,   // [CLS][N][HD]
             float* __restrict__ attn) {  // [N][CLS]
  extern __shared__ char smem_raw[];
  float* xbuf = (float*)smem_raw;        // [MT][FEAT] x tile, later V tile (128 KB)
  float* sbuf = xbuf + MT * FEAT;        // [MT][SC] scores, later alpha*sharp (20 KB)

  const int tid  = threadIdx.x;
  const int n0   = blockIdx.x * MT;
  const int wave = tid >> 5;
  const int lane = tid & 31;
  const int hi   = lane >> 4;            // lane half (0/1)
  const int l15  = lane & 15;

  // ---- stage x tile -> LDS via async global->LDS copies (ASYNCcnt) ----
  // LDS aperture: flat addr low 32 bits == LDS byte offset (ISA 10.2).
  {
    const uint32_t lds_base = (uint32_t)(uintptr_t)xbuf;
    #pragma unroll
    for (int i = 0; i < (MT * FEAT / 4) / BLK; ++i) {   // 16 iters
      const int q   = i * BLK + tid;                    // float4 index
      const int row = q >> 7;                           // FEAT/4 = 128 per row
      const int c4  = q & 127;
      if (n0 + row < N_ROWS) {
        const uint32_t ldst = lds_base + (uint32_t)(row * FEAT + c4 * 4) * 4u;
        const float* g = x + (size_t)(n0 + row) * FEAT + c4 * 4;
        asm volatile("global_load_async_to_lds_b128 %0, %1, off"
                     :: "v"(ldst), "v"(g) : "memory");
      } else {
        v4f z = 0.f;
        *(v4f*)(xbuf + row * FEAT + c4 * 4) = z;        // zero-pad tail rows
      }
    }
    asm volatile("s_wait_asynccnt 0x0" ::: "memory");   // drain this wave
  }
  __syncthreads();

  // ---- GEMM1: scores[MT][80] = xtile @ Wqk + bqk (wmma f32 16x16x4) ----
  // 20 (mi,nj) subtile jobs; wave handles {wave} and (waves 0..3) {wave+16}.
  auto gemm1_job = [&](int job) {
    const int mi = job / 5, nj = job % 5;
    const int col = nj * 16 + l15;
    v8f acc;
    const float bi = bqk[col];
    #pragma unroll
    for (int r = 0; r < 8; ++r) acc[r] = bi;
    const float* arow = xbuf + (mi * 16 + l15) * FEAT + hi * 2;
    for (int kc = 0; kc < KSTEPS; ++kc) {
      const v2f a = *(const v2f*)(arow + kc * 4);
      const int kb = kc * 4 + 2 * hi;
      v2f b;
      b.x = Wqk[(kb + 0) * SC + col];
      b.y = Wqk[(kb + 1) * SC + col];
      acc = __builtin_amdgcn_wmma_f32_16x16x4_f32(false, a, false, b,
                                                  (short)0, acc, false, false);
    }
    #pragma unroll
    for (int r = 0; r < 8; ++r)
      sbuf[(mi * 16 + r + 8 * hi) * SC + col] = acc[r];
  };
  gemm1_job(wave);            // wave-uniform control: EXEC stays all-ones
  if (wave < 4) gemm1_job(wave + 16);

  // ---- GEMM2: V[MT][512] = xtile @ Wv + bv, accumulate in registers ----
  const int mi2 = wave & 3;            // M-subtile
  const int njb = (wave >> 2) * 8;     // 8 N-subtiles per wave
  v8f vacc[8];
  #pragma unroll
  for (int j = 0; j < 8; ++j) {
    const float bvv = bv[(njb + j) * 16 + l15];
    #pragma unroll
    for (int r = 0; r < 8; ++r) vacc[j][r] = bvv;
  }
  {
    const float* arow = xbuf + (mi2 * 16 + l15) * FEAT + hi * 2;
    for (int kc = 0; kc < KSTEPS; ++kc) {
      const v2f a = *(const v2f*)(arow + kc * 4);            // shared A frag
      const int kb = kc * 4 + 2 * hi;
      const float* w0 = Wv + (size_t)kb * HD;
      #pragma unroll
      for (int j = 0; j < 8; ++j) {
        const int col = (njb + j) * 16 + l15;
        v2f b;
        b.x = w0[col];
        b.y = w0[HD + col];
        vacc[j] = __builtin_amdgcn_wmma_f32_16x16x4_f32(false, a, false, b,
                                                        (short)0, vacc[j],
                                                        false, false);
      }
    }
  }
  __syncthreads();   // all xbuf reads + all score writes done

  // ---- write V over the x tile (transpose through LDS for b128 reads) ----
  #pragma unroll
  for (int j = 0; j < 8; ++j)
    #pragma unroll
    for (int r = 0; r < 8; ++r)
      xbuf[(mi2 * 16 + r + 8 * hi) * FEAT + (njb + j) * 16 + l15] = vacc[j][r];

  // ---- per-(row, head) softmax / mask / entropy / sharp; attn via shfl ----
  {
    const int row = tid >> 3;          // 0..63
    const int h   = tid & 7;
    float* slot = sbuf + row * SC + h * CLS;   // this thread's 10 scores
    float sc[CLS], alpha[CLS];
    float m = -1e30f;
    #pragma unroll
    for (int c = 0; c < CLS; ++c) { sc[c] = slot[c]; m = fmaxf(m, sc[c]); }
    float sum = 0.f;
    #pragma unroll
    for (int c = 0; c < CLS; ++c) { sc[c] = expf(sc[c] - m); sum += sc[c]; }
    const float inv = 1.f / sum;
    const int n = n0 + row;
    const bool mid = (n < N_ROWS) ? (mid_mask[n] != 0) : false;
    float ent = 0.f;
    #pragma unroll
    for (int c = 0; c < CLS; ++c) {
      const float a = mid ? (1.f / CLS) : sc[c] * inv;
      alpha[c] = a;
      ent = fmaf(a, logf(a + 1e-6f), ent);     // ent = -entropy
    }
    const float sharp = expf(ent);             // exp(-entropy)
    // mean over heads: 8 consecutive lanes (same row) inside one wave32
    #pragma unroll
    for (int c = 0; c < CLS; ++c) {
      float a = alpha[c];
      a += __shfl_xor(a, 1, 32);
      a += __shfl_xor(a, 2, 32);
      a += __shfl_xor(a, 4, 32);
      if (h == 0 && n < N_ROWS) attn[(size_t)n * CLS + c] = a * 0.125f;
    }
    // overwrite own score slots with alpha * sharp (private slots: no race)
    #pragma unroll
    for (int c = 0; c < CLS; ++c) slot[c] = alpha[c] * sharp;
  }
  __syncthreads();

  // ---- expansion: track[c][n][hd] = V[n][hd] * scale[n][h(hd)][c] ----
  // 64 rows * 10 classes * 128 float4 = 81920 b128 NT stores / block.
  // Nontemporal: output is write-once, keep L2 for Wv/Wqk reuse instead.
  #pragma unroll 4
  for (int it = 0; it < (MT * CLS * (HD / 4)) / BLK; ++it) {  // 160 iters
    const int idx = it * BLK + tid;
    const int c4  = idx & 127;           // float4 column
    const int rc  = idx >> 7;
    const int row = rc & (MT - 1);
    const int c   = rc >> 6;
    const int n   = n0 + row;
    if (n < N_ROWS) {
      v4f v = *(const v4f*)(xbuf + row * FEAT + c4 * 4);
      const float s = sbuf[row * SC + (c4 >> 4) * CLS + c];  // h = c4>>4
      v *= s;
      __builtin_nontemporal_store(
          v, (v4f*)(track + ((size_t)c * N_ROWS + n) * HD + c4 * 4));
    }
  }
}

// ---------------------------------------------------------------------------
extern "C" void kernel_launch(void* const* d_in, const int* in_sizes, int n_in,
                              void* d_out, int out_size, void* d_ws, size_t ws_size,
                              hipStream_t stream) {
  const float* x     = (const float*)d_in[0];
  const float* proto = (const float*)d_in[1];
  // d_in[2] = pre_emb (unused by the reference forward)
  const unsigned char* mid_mask = (const unsigned char*)d_in[3];  // jnp bool_
  const float* Wv = (const float*)d_in[4];
  const float* bv = (const float*)d_in[5];
  const float* Wq = (const float*)d_in[6];
  const float* bq = (const float*)d_in[7];
  const float* Wk = (const float*)d_in[8];
  const float* bk = (const float*)d_in[9];
  // d_in[10] = n_heads (== HEADS, compile-time constant here)

  float* Wqk = (float*)d_ws;                    // [FEAT][SC]
  float* bqk = Wqk + FEAT * SC;                 // [SC]
  float* Kws = bqk + SC;                        // [CLS][HD]

  float* track = (float*)d_out;                             // [CLS][N][HD]
  float* attn  = track + (size_t)CLS * N_ROWS * HD;         // [N][CLS]

  prep_k_kernel<<<CLS, HD, 0, stream>>>(proto, Wk, bk, Kws);
  prep_wqk_kernel<<<FEAT / 8, BLK, 0, stream>>>(Wq, bq, Kws, Wqk, bqk);

  const int grid = (N_ROWS + MT - 1) / MT;                  // 313
  const size_t smem = (size_t)(MT * FEAT + MT * SC) * sizeof(float);  // 151.5 KB
  track_kernel<<<grid, BLK, smem, stream>>>(x, mid_mask, Wv, bv, Wqk, bqk,
                                            track, attn);
}